// CryptoGNN_8856222564958
// MI455X (gfx1250) — compile-verified
//
#include <hip/hip_runtime.h>

typedef __attribute__((ext_vector_type(2)))  float v2f;
typedef __attribute__((ext_vector_type(8)))  float v8f;

// ---------------------------------------------------------------------------
// Degree / dinv
// ---------------------------------------------------------------------------
__global__ void gcn_init_deg(float* __restrict__ deg, float* __restrict__ pooled, int N) {
  int t = blockIdx.x * blockDim.x + threadIdx.x;
  if (t < 64) pooled[t] = 0.0f;
  if (t < N) deg[t] = 1.0f;              // self-loop
}

__global__ void gcn_count_deg(const int* __restrict__ dst, float* __restrict__ deg, int E) {
  int e = blockIdx.x * blockDim.x + threadIdx.x;
  if (e < E) atomicAdd(&deg[dst[e]], 1.0f);
}

__global__ void gcn_rsqrt(float* __restrict__ deg, int N) {
  int t = blockIdx.x * blockDim.x + threadIdx.x;
  if (t < N) deg[t] = rsqrtf(deg[t]);
}

// ---------------------------------------------------------------------------
// Dense transform: Hout[N x 64] = (RELU? relu(X) : X)[N x K] @ W[K x 64]
// One wave per 16x16 output tile, V_WMMA_F32_16X16X4_F32, K compile-time.
// ---------------------------------------------------------------------------
template <int K, bool RELU>
__global__ void gcn_gemm_wmma(const float* __restrict__ X, const float* __restrict__ W,
                              float* __restrict__ Hout, int N) {
  const int HD = 64;
  const int lane = threadIdx.x & 31;
  const int wave = (blockIdx.x * blockDim.x + threadIdx.x) >> 5;
  const int rowTiles = (N + 15) >> 4;
  const int totalTiles = rowTiles * (HD / 16);
  if (wave >= totalTiles) return;          // wave-uniform exit, EXEC stays full

  const int tm = wave >> 2;                // HD/16 == 4 col tiles
  const int tn = wave & 3;
  const int row0 = tm << 4, col0 = tn << 4;
  int mrow = row0 + (lane & 15);
  mrow = (mrow < N) ? mrow : (N - 1);      // clamp: select, no divergence
  const int n  = col0 + (lane & 15);
  const int kh = (lane >> 4) << 1;         // f32 A 16x4 frag: lanes 0-15 K=0,1 ; 16-31 K=2,3

  const float* __restrict__ xrow = X + (size_t)mrow * K;

  v8f acc = {};
#pragma unroll
  for (int k0 = 0; k0 < K; k0 += 4) {
    const int ka = k0 + kh;
    v2f a, b;
    if constexpr ((K & 3) == 0) {
      // fully in-range: aligned pair load for A, two b32 for B
      a = *(const v2f*)(xrow + ka);
      if (RELU) { a.x = fmaxf(a.x, 0.0f); a.y = fmaxf(a.y, 0.0f); }
      b.x = W[(size_t)ka * HD + n];
      b.y = W[(size_t)(ka + 1) * HD + n];
    } else {
      // tail-capable step: clamped unconditional loads + cndmask zeroing of A
      const int kc0 = (ka     < K) ? ka     : (K - 1);
      const int kc1 = (ka + 1 < K) ? ka + 1 : (K - 1);
      float x0 = xrow[kc0];
      float x1 = xrow[kc1];
      if (RELU) { x0 = fmaxf(x0, 0.0f); x1 = fmaxf(x1, 0.0f); }
      a.x = (ka     < K) ? x0 : 0.0f;      // zero A => zero product
      a.y = (ka + 1 < K) ? x1 : 0.0f;
      b.x = W[(size_t)kc0 * HD + n];
      b.y = W[(size_t)kc1 * HD + n];
    }
    acc = __builtin_amdgcn_wmma_f32_16x16x4_f32(false, a, false, b,
                                                (short)0, acc, false, false);
  }

  // C/D layout: lanes 0-15 -> M = v ; lanes 16-31 -> M = v + 8
  const int rbase = row0 + ((lane >> 4) << 3);
  const int c = col0 + (lane & 15);
  float* __restrict__ outp = Hout + (size_t)rbase * HD + c;
  if (row0 + 16 <= N) {
    // full tile (wave-uniform): 8 unguarded stores off one base, imm offsets
#pragma unroll
    for (int v = 0; v < 8; ++v) outp[(size_t)v * HD] = acc[v];
  } else {
    // partial tile tail
#pragma unroll
    for (int v = 0; v < 8; ++v)
      if (rbase + v < N) outp[(size_t)v * HD] = acc[v];
  }
}

// ---------------------------------------------------------------------------
// out[i][f] = bias[f] + h[i][f] * dinv[i]^2   (self-loop folded in)
// ---------------------------------------------------------------------------
__global__ void gcn_init_out(const float* __restrict__ h, const float* __restrict__ dinv,
                             const float* __restrict__ bias, float* __restrict__ out, int N) {
  long long t = (long long)blockIdx.x * blockDim.x + threadIdx.x;
  if (t >= (long long)N * 64) return;
  const int i = (int)(t >> 6);
  const int f = (int)(t & 63);
  const float di = dinv[i];
  out[t] = bias[f] + h[t] * di * di;
}

// ---------------------------------------------------------------------------
// Edge scatter: out[d] += h[s] * dinv[s]*dinv[d].  16 threads per edge,
// float4 gather + 4 global f32 atomics each.
// ---------------------------------------------------------------------------
__global__ void gcn_scatter(const int* __restrict__ src, const int* __restrict__ dst,
                            const float* __restrict__ dinv, const float* __restrict__ h,
                            float* __restrict__ out, int E) {
  const int HD = 64;
  long long gtid = (long long)blockIdx.x * blockDim.x + threadIdx.x;
  const int e  = (int)(gtid >> 4);
  const int f4 = (int)(gtid & 15) << 2;
  if (e >= E) return;
  const int s = src[e];
  const int d = dst[e];
  const float norm = dinv[s] * dinv[d];
  const float4 hv = *(const float4*)(h + (size_t)s * HD + f4);
  float* op = out + (size_t)d * HD + f4;
  atomicAdd(op + 0, hv.x * norm);
  atomicAdd(op + 1, hv.y * norm);
  atomicAdd(op + 2, hv.z * norm);
  atomicAdd(op + 3, hv.w * norm);
}

// ---------------------------------------------------------------------------
// Column sums of h[N x 64] into pooled[64]
// ---------------------------------------------------------------------------
__global__ void gcn_pool(const float* __restrict__ h, float* __restrict__ pooled, int N) {
  const int t = blockIdx.x * blockDim.x + threadIdx.x;
  const int f = t & 63;
  const int g = t >> 6;
  const int stride = (gridDim.x * blockDim.x) >> 6;
  float s = 0.0f;
  for (int i = g; i < N; i += stride) s += h[(size_t)i * 64 + f];
  atomicAdd(&pooled[f], s);
}

// ---------------------------------------------------------------------------
// out[0] = (pooled/N) . fcW + fcb   (single block, 64 threads)
// ---------------------------------------------------------------------------
__global__ void gcn_final(const float* __restrict__ pooled, const float* __restrict__ fcW,
                          const float* __restrict__ fcb, float* __restrict__ out, int N) {
  __shared__ float s[64];
  const int f = threadIdx.x;
  s[f] = (pooled[f] / (float)N) * fcW[f];
  __syncthreads();
  for (int off = 32; off > 0; off >>= 1) {
    if (f < off) s[f] += s[f + off];
    __syncthreads();
  }
  if (f == 0) out[0] = s[0] + fcb[0];
}

// ---------------------------------------------------------------------------
extern "C" void kernel_launch(void* const* d_in, const int* in_sizes, int n_in,
                              void* d_out, int out_size, void* d_ws, size_t ws_size,
                              hipStream_t stream) {
  const float* x   = (const float*)d_in[0];
  const int*   ei  = (const int*)  d_in[1];
  const float* W1  = (const float*)d_in[2];
  const float* b1  = (const float*)d_in[3];
  const float* W2  = (const float*)d_in[4];
  const float* b2  = (const float*)d_in[5];
  const float* W3  = (const float*)d_in[6];
  const float* b3  = (const float*)d_in[7];
  const float* fcW = (const float*)d_in[8];
  const float* fcb = (const float*)d_in[9];
  float* out = (float*)d_out;

  const int FIN = 10, HD = 64;
  const int N = in_sizes[0] / FIN;
  const int E = in_sizes[1] / 2;
  const int* src = ei;
  const int* dst = ei + E;

  // Workspace layout (f32): dinv[N(pad16)] | pooled[64] | bufA[N*64] | bufB[N*64]
  float* ws = (float*)d_ws;
  size_t o = 0;
  float* dinv   = ws + o; o += (size_t)((N + 15) & ~15);
  float* pooled = ws + o; o += 64;
  float* bufA   = ws + o; o += (size_t)N * HD;   // GEMM output h
  float* bufB   = ws + o;                        // aggregated output

  const int BLK = 256;
  const int gN   = (N + BLK - 1) / BLK;
  const int gE   = (E + BLK - 1) / BLK;
  const int gNH  = (int)(((long long)N * HD + BLK - 1) / BLK);
  const int gSct = (int)(((long long)E * 16 + BLK - 1) / BLK);
  const int waves = ((N + 15) >> 4) * (HD / 16);
  const int gGemm = (waves * 32 + BLK - 1) / BLK;

  // degrees
  gcn_init_deg <<<gN, BLK, 0, stream>>>(dinv, pooled, N);
  gcn_count_deg<<<gE, BLK, 0, stream>>>(dst, dinv, E);
  gcn_rsqrt    <<<gN, BLK, 0, stream>>>(dinv, N);

  // layer 1 (raw x, K=10)
  gcn_gemm_wmma<10, false><<<gGemm, BLK, 0, stream>>>(x, W1, bufA, N);
  gcn_init_out <<<gNH, BLK, 0, stream>>>(bufA, dinv, b1, bufB, N);
  gcn_scatter  <<<gSct, BLK, 0, stream>>>(src, dst, dinv, bufA, bufB, E);

  // layer 2 (relu fused into GEMM A-load, K=64)
  gcn_gemm_wmma<64, true> <<<gGemm, BLK, 0, stream>>>(bufB, W2, bufA, N);
  gcn_init_out <<<gNH, BLK, 0, stream>>>(bufA, dinv, b2, bufB, N);
  gcn_scatter  <<<gSct, BLK, 0, stream>>>(src, dst, dinv, bufA, bufB, E);

  // layer 3 (relu fused, K=64, no output relu)
  gcn_gemm_wmma<64, true> <<<gGemm, BLK, 0, stream>>>(bufB, W3, bufA, N);
  gcn_init_out <<<gNH, BLK, 0, stream>>>(bufA, dinv, b3, bufB, N);
  gcn_scatter  <<<gSct, BLK, 0, stream>>>(src, dst, dinv, bufA, bufB, E);

  // mean pool + FC
  gcn_pool <<<256, BLK, 0, stream>>>(bufB, pooled, N);
  gcn_final<<<1, 64, 0, stream>>>(pooled, fcW, fcb, out, N);
}